// AnotherSCLHead_83253646066146
// MI455X (gfx1250) — compile-verified
//
#include <hip/hip_runtime.h>
#include <math.h>

// CDNA5 / gfx1250, wave32.
typedef float v2f __attribute__((ext_vector_type(2)));
typedef float v8f __attribute__((ext_vector_type(8)));

#define NROWS   8192      // 2N
#define HALF    4096      // N
#define DIM     256       // feature dim
#define TILE    16
#define RPB     32        // rows per block (two 16-row A panels share each B tile)
#define NCT     (NROWS / TILE)   // 512 column tiles
#define WAVES   8
#define THREADS 256
#define LDA     260       // padded LDS row stride (floats): 260 % 64 == 4 -> conflict-free
#define POS_CAP 2048
#define INV_T   10.0f     // 1 / TEMPERATURE

__global__ void __launch_bounds__(64) scl_zero_out(float* out) {
    if (threadIdx.x == 0) out[0] = 0.0f;
}

__device__ __forceinline__ const float* frow(const float* features, int r) {
    // f[r] = features[r & 4095, r >> 12, :]   (concat of the two views)
    return features + (size_t)(r & (HALF - 1)) * (2 * DIM) + ((r >> 12) ? DIM : 0);
}

__global__ void __launch_bounds__(THREADS)
scl_fused_kernel(const float* __restrict__ features,
                 const int*   __restrict__ label,
                 float*       __restrict__ out)
{
    // 32x256 A panel (2 x 16-row panels). After the GEMM phase this buffer is
    // reused as the (wave, row, lane16) LSE-reduction scratch: needs
    // 2 * 8*32*16 = 8192 floats <= 32*260 = 8320 floats.
    __shared__ float A_lds[RPB * LDA];
    __shared__ float lse_lds[RPB];
    __shared__ int   lab_row[RPB];
    __shared__ int   cnt_lds[RPB];
    __shared__ int   pos_n;
    __shared__ int   pos_row[POS_CAP];
    __shared__ float pos_val[POS_CAP];
    __shared__ float partials[THREADS];

    const int tid  = threadIdx.x;
    const int lane = tid & 31;
    const int wave = tid >> 5;
    const int R0   = blockIdx.x * RPB;

    // ---- stage the 32x256 A panel into LDS (padded stride) ----
    for (int i = tid; i < RPB * DIM; i += THREADS) {
        const int m = i >> 8;          // row in block
        const int k = i & (DIM - 1);   // feature index
        A_lds[m * LDA + k] = frow(features, R0 + m)[k];
    }
    if (tid < RPB) {
        lab_row[tid] = label[(R0 + tid) & (HALF - 1)];
        cnt_lds[tid] = 0;
    }
    if (tid == 0) pos_n = 0;
    __syncthreads();

    const int cloc  = lane & 15;   // A: M-row / B: N-col within tile
    const int khalf = lane >> 4;   // selects K pair {0,1} vs {2,3}
    const int rbase = khalf * 8;   // C/D rows held by this lane: rbase..rbase+7

    // online (max, sum) over NEGATIVES: slot index = r_local = 16*panel + rbase + j
    float run_m[16], run_s[16];
    #pragma unroll
    for (int j = 0; j < 16; ++j) { run_m[j] = -INFINITY; run_s[j] = 0.0f; }

    const float* aptr0 = &A_lds[cloc * LDA + 2 * khalf];               // panel 0
    const float* aptr1 = &A_lds[(TILE + cloc) * LDA + 2 * khalf];      // panel 1

    // ---- stream all column tiles: wave w handles tiles w, w+8, ... ----
    for (int ct = wave; ct < NCT; ct += WAVES) {
        const int C = ct * TILE + cloc;
        const float* bptr = frow(features, C) + 2 * khalf;

        v8f acc0 = {};
        v8f acc1 = {};
        #pragma unroll 8
        for (int kk = 0; kk < DIM / 4; ++kk) {
            v2f b  = *(const v2f*)(bptr  + kk * 4);  // global_load_b64 (L2-resident), shared by both panels
            v2f a0 = *(const v2f*)(aptr0 + kk * 4);  // ds_load_b64
            v2f a1 = *(const v2f*)(aptr1 + kk * 4);  // ds_load_b64
            acc0 = __builtin_amdgcn_wmma_f32_16x16x4_f32(
                     false, a0, false, b, (short)0, acc0, false, false);
            acc1 = __builtin_amdgcn_wmma_f32_16x16x4_f32(
                     false, a1, false, b, (short)0, acc1, false, false);
        }

        const int labC = label[C & (HALF - 1)];
        #pragma unroll
        for (int p = 0; p < 2; ++p) {
            #pragma unroll
            for (int j = 0; j < 8; ++j) {
                const int r_local = p * TILE + rbase + j;
                const int R = R0 + r_local;
                if (C == R) continue;                 // drop self-similarity
                const float s = (p ? acc1[j] : acc0[j]) * INV_T;
                if (labC == lab_row[r_local]) {       // positive pair: stash logit
                    const int idx = atomicAdd(&pos_n, 1);
                    if (idx < POS_CAP) { pos_row[idx] = r_local; pos_val[idx] = s; }
                    atomicAdd(&cnt_lds[r_local], 1);
                } else {                              // negative: online logsumexp
                    const float mo = run_m[r_local & 15 | (p << 4)];  // == r_local
                    const float mn = fmaxf(mo, s);
                    run_s[r_local] = run_s[r_local] * __expf(mo - mn) + __expf(s - mn);
                    run_m[r_local] = mn;
                }
            }
        }
    }

    // ---- A panel is dead: reuse its LDS as reduction scratch ----
    __syncthreads();   // all waves done reading A_lds
    float* red_m = A_lds;                      // [WAVES][RPB][16] = 4096 floats
    float* red_s = A_lds + WAVES * RPB * 16;   // another 4096 floats

    #pragma unroll
    for (int p = 0; p < 2; ++p) {
        #pragma unroll
        for (int j = 0; j < 8; ++j) {
            const int r_local = p * TILE + rbase + j;
            const int slot = (wave * RPB + r_local) * 16 + cloc;
            red_m[slot] = run_m[r_local];
            red_s[slot] = run_s[r_local];
        }
    }
    __syncthreads();

    // ---- per-row LSE over negatives: 32 threads, one row each ----
    if (tid < RPB) {
        float m = -INFINITY, ssum = 0.0f;
        for (int w = 0; w < WAVES; ++w) {
            for (int l = 0; l < 16; ++l) {
                const int slot = (w * RPB + tid) * 16 + l;
                const float m2 = red_m[slot];
                if (m2 == -INFINITY) continue;    // empty partial
                const float s2 = red_s[slot];
                const float mn = fmaxf(m, m2);
                ssum = ssum * __expf(m - mn) + s2 * __expf(m2 - mn);
                m = mn;
            }
        }
        lse_lds[tid] = m + __logf(ssum);
    }
    __syncthreads();

    // ---- positives: loss = softplus(lse_neg - p), averaged per row then globally ----
    float partial = 0.0f;
    const int npos = pos_n < POS_CAP ? pos_n : POS_CAP;
    for (int i = tid; i < npos; i += THREADS) {
        const int   r = pos_row[i];
        const float p = pos_val[i];
        const float d = lse_lds[r] - p;           // logaddexp(p, L) - p = softplus(L - p)
        const float term = (d > 0.0f) ? d + log1pf(__expf(-d)) : log1pf(__expf(d));
        partial += term / (float)cnt_lds[r];
    }
    partials[tid] = partial;
    __syncthreads();
    for (int off = THREADS / 2; off > 0; off >>= 1) {
        if (tid < off) partials[tid] += partials[tid + off];
        __syncthreads();
    }
    if (tid == 0) atomicAdd(out, partials[0] * (1.0f / (float)NROWS));
}

extern "C" void kernel_launch(void* const* d_in, const int* in_sizes, int n_in,
                              void* d_out, int out_size, void* d_ws, size_t ws_size,
                              hipStream_t stream) {
    const float* features = (const float*)d_in[0];   // [4096, 2, 256] fp32
    const int*   label    = (const int*)d_in[1];     // [4096] int32
    float*       out      = (float*)d_out;           // scalar loss

    scl_zero_out<<<1, 64, 0, stream>>>(out);
    scl_fused_kernel<<<NROWS / RPB, THREADS, 0, stream>>>(features, label, out);
}